// AcousticPhysicsEngine_86792699117905
// MI455X (gfx1250) — compile-verified
//
#include <hip/hip_runtime.h>

#ifndef __has_builtin
#define __has_builtin(x) 0
#endif

#define COLS_TOTAL 65536        // 256*256 field elements = 256 KB in LDS
#define BLOCK_MAIN 1024         // 32 waves per WGP (wave32)
#define GRID_MAIN  512

// Pointer types matching the async-to-LDS builtin signature:
// __builtin_amdgcn_global_load_async_to_lds_b128(v4i AS(1)*, v4i AS(3)*, Ii, Ii)
typedef int v4i __attribute__((ext_vector_type(4)));
typedef __attribute__((address_space(1))) v4i gv4i;   // global
typedef __attribute__((address_space(3))) v4i lv4i;   // LDS

// flat[k] = field_map[k & 255][k >> 8] -> row-major offset ((k&255)<<8)|(k>>8).
// For k < 65536 this is a 16-bit byte swap: one v_perm_b32.
__device__ __forceinline__ int transpose_idx(int c) {
#if __has_builtin(__builtin_amdgcn_perm)
  // out.b0 = c.b1 (sel 1), out.b1 = c.b0 (sel 0), out.b2 = out.b3 = 0 (sel 0x0C)
  return (int)__builtin_amdgcn_perm(0u, (unsigned)c, 0x0C0C0001u);
#else
  return ((c & 255) << 8) | ((unsigned)c >> 8);
#endif
}

__global__ __launch_bounds__(256) void zero_kernel(float4* __restrict__ out, int n4) {
  int i = blockIdx.x * 256 + threadIdx.x;
  if (i < n4) out[i] = make_float4(0.f, 0.f, 0.f, 0.f);   // global_store_b128
}

__global__ __launch_bounds__(BLOCK_MAIN) void coo_spmv_kernel(
    const float* __restrict__ field,
    const int*   __restrict__ idx_row,
    const int*   __restrict__ idx_col,
    const float* __restrict__ vals,
    float*       __restrict__ out,
    int nnz)
{
  // Entire dense vector lives in LDS (256 KB of the WGP's 320 KB).
  __shared__ float lds_field[COLS_TOTAL];

  // ---- Stage field map into LDS -------------------------------------------
#if __has_builtin(__builtin_amdgcn_global_load_async_to_lds_b128)
  // CDNA5 async copy: memory -> LDS, 16B per lane, tracked by ASYNCcnt.
  for (int base = threadIdx.x * 4; base < COLS_TOTAL; base += BLOCK_MAIN * 4) {
    __builtin_amdgcn_global_load_async_to_lds_b128(
        (gv4i*)(field + base),
        (lv4i*)(&lds_field[base]),
        /*offset=*/0, /*cpol=*/0);
  }
#if __has_builtin(__builtin_amdgcn_s_wait_asynccnt)
  __builtin_amdgcn_s_wait_asynccnt(0);
#else
  asm volatile("s_wait_asynccnt 0" ::: "memory");
#endif
#else
  // Fallback: 128-bit global loads + 128-bit ds stores.
  {
    const float4* __restrict__ src = (const float4*)field;
    float4* dst = (float4*)lds_field;
    for (int i = threadIdx.x; i < COLS_TOTAL / 4; i += BLOCK_MAIN) dst[i] = src[i];
  }
#endif
  __syncthreads();

  // ---- Stream the COO triples, gather from LDS, scatter-add to L2 --------
  const int nq     = nnz >> 2;               // 128-bit vectorized quads
  const int stride = gridDim.x * BLOCK_MAIN;
  const int4*   __restrict__ col4 = (const int4*)idx_col;
  const int4*   __restrict__ row4 = (const int4*)idx_row;
  const float4* __restrict__ val4 = (const float4*)vals;

  for (int q = blockIdx.x * BLOCK_MAIN + threadIdx.x; q < nq; q += stride) {
    int4   c = col4[q];          // global_load_b128
    int4   r = row4[q];          // global_load_b128
    float4 v = val4[q];          // global_load_b128

    int p0 = transpose_idx(c.x);  // v_perm_b32 each
    int p1 = transpose_idx(c.y);
    int p2 = transpose_idx(c.z);
    int p3 = transpose_idx(c.w);

    float f0 = lds_field[p0];    // ds_load_b32 gathers (LDS-resident vector)
    float f1 = lds_field[p1];
    float f2 = lds_field[p2];
    float f3 = lds_field[p3];

    // fire-and-forget float atomics (non-returning -> STOREcnt only)
    atomicAdd(&out[r.x], v.x * f0);
    atomicAdd(&out[r.y], v.y * f1);
    atomicAdd(&out[r.z], v.z * f2);
    atomicAdd(&out[r.w], v.w * f3);
  }

  // scalar tail (nnz not divisible by 4)
  for (int i = (nq << 2) + blockIdx.x * BLOCK_MAIN + threadIdx.x; i < nnz; i += stride) {
    atomicAdd(&out[idx_row[i]], vals[i] * lds_field[transpose_idx(idx_col[i])]);
  }
}

extern "C" void kernel_launch(void* const* d_in, const int* in_sizes, int n_in,
                              void* d_out, int out_size, void* d_ws, size_t ws_size,
                              hipStream_t stream) {
  const float* field   = (const float*)d_in[0];  // [256*256] f32
  const int*   idx_row = (const int*)d_in[1];    // [NNZ] int
  const int*   idx_col = (const int*)d_in[2];    // [NNZ] int
  const float* vals    = (const float*)d_in[3];  // [NNZ] f32
  float* out = (float*)d_out;                    // [131072] f32
  const int nnz = in_sizes[1];

  // out_size = 131072 (divisible by 4) -> vectorized zeroing
  const int n4 = out_size >> 2;
  zero_kernel<<<(n4 + 255) / 256, 256, 0, stream>>>((float4*)out, n4);
  coo_spmv_kernel<<<GRID_MAIN, BLOCK_MAIN, 0, stream>>>(field, idx_row, idx_col, vals,
                                                        out, nnz);
}